// GIN_14680198218264
// MI455X (gfx1250) — compile-verified
//
#include <hip/hip_runtime.h>
#include <hip/hip_bf16.h>

// ---------------------------------------------------------------------------
// GIN forward on gfx1250 (MI455X).
//   - fp32 WMMA (V_WMMA_F32_16X16X4_F32) for all dense layers (exact f32)
//   - weight tiles staged in LDS via async global->LDS copies (ASYNCcnt)
//   - native fp32 global atomics for edge scatter / BN stats / pooling
// ---------------------------------------------------------------------------

typedef __attribute__((ext_vector_type(2))) float v2f;
typedef __attribute__((ext_vector_type(8))) float v8f;

#define IN_C    128
#define HID     256
#define OUT_C   64
#define NGRAPH  256
#define BN_EPS  1e-5f
#define KT      64          // K-tile staged in LDS per buffer

__device__ __forceinline__ void atomAddF(float* p, float v) {
#if defined(__HIP_DEVICE_COMPILE__)
    unsafeAtomicAdd(p, v);           // native global_atomic_add_f32
#else
    atomicAdd(p, v);
#endif
}

// ---------------------------------------------------------------- utilities

__global__ void zero_f32(float* __restrict__ p, long long n) {
    long long i = (long long)blockIdx.x * blockDim.x + threadIdx.x;
    if (i < n) p[i] = 0.0f;
}

__global__ void copy_f32v4(float* __restrict__ dst, const float* __restrict__ src,
                           long long n4) {
    long long i = (long long)blockIdx.x * blockDim.x + threadIdx.x;
    if (i < n4) {
        ((float4*)dst)[i] = ((const float4*)src)[i];
    }
}

// ------------------------------------------------------------ edge scatter
// hsum[dst] += h[src]  (hsum pre-initialized with h => (1+eps)*x + sum)

__global__ void scatter_edges(float* __restrict__ hsum, const float* __restrict__ h,
                              const long long* __restrict__ src,
                              const long long* __restrict__ dst,
                              long long E, int C) {
    const int vecs = C >> 2;
    long long tid = (long long)blockIdx.x * blockDim.x + threadIdx.x;
    if (tid >= E * (long long)vecs) return;
    long long e  = tid / vecs;
    int       c4 = (int)(tid % vecs) << 2;
    long long s = src[e];
    long long d = dst[e];
    float4 v = *(const float4*)(h + s * (long long)C + c4);
    float* out = hsum + d * (long long)C + c4;
    atomAddF(out + 0, v.x);
    atomAddF(out + 1, v.y);
    atomAddF(out + 2, v.z);
    atomAddF(out + 3, v.w);
}

// ---------------------------------------------------------------- WMMA GEMM
// C[M,N] = op(A[M,K] @ B[K,N] + bias[N]),  op = relu if relu!=0.
// Block: 256 threads = 8 waves. Block tile 128x64; wave tile 16x64.
// B is staged in LDS (double-buffered KTx64 tiles) via async global->LDS
// copies; compute waits with s_wait_asynccnt + workgroup barrier.
// f32 WMMA fragment layout (ISA 7.12.2, wave32):
//   A (16x4): lane -> M = lane&15 ; VGPR{0,1} -> K = k0 + 2*(lane>>4) + {0,1}
//   B (4x16): lane -> N = lane&15 ; VGPR{0,1} -> K = k0 + 2*(lane>>4) + {0,1}
//   C (16x16): VGPR r -> row r + 8*(lane>>4), col = lane&15

__global__ void __launch_bounds__(256)
wmma_gemm_f32(const float* __restrict__ A, const float* __restrict__ B,
              const float* __restrict__ bias, float* __restrict__ C,
              int M, int K, int N, int relu) {
    __shared__ float ldsB[2][KT * 64];               // 2 x 16 KB

    const int lane  = threadIdx.x & 31;
    const int wave  = threadIdx.x >> 5;
    const int nIdx  = lane & 15;   // M index for A frag, N index for B/C frags
    const int khalf = lane >> 4;   // 0 or 1

    const int rowBase = blockIdx.x * 128 + wave * 16;
    const int colBase = blockIdx.y * 64;

    // Clamp A row so every lane issues a valid load: EXEC must stay all-ones
    // through the WMMA loop. Clamped (duplicate) rows are never stored.
    int aRow = rowBase + nIdx;
    if (aRow >= M) aRow = M - 1;
    const float* __restrict__ Arow = A + (long long)aRow * K;

    // Cooperative async copy of B[kt..kt+KT) x [colBase..colBase+64) into LDS.
    // KT*64 = 4096 floats = 1024 xfers of 16B; 256 threads x 4.
    auto loadBtile = [&](int kt, int buf) {
#pragma unroll
        for (int i = 0; i < 4; ++i) {
            const int f4 = i * 256 + threadIdx.x;    // 0..1023
            const int kk = f4 >> 4;                  // 0..63
            const int n4 = (f4 & 15) << 2;           // 0..60
            const float* g = B + (long long)(kt + kk) * N + colBase + n4;
            const unsigned l = (unsigned)(size_t)&ldsB[buf][kk * 64 + n4];
            asm volatile("global_load_async_to_lds_b128 %0, %1, off"
                         :: "v"(l), "v"(g) : "memory");
        }
    };

    v8f acc0 = {}, acc1 = {}, acc2 = {}, acc3 = {};

    loadBtile(0, 0);
    for (int kt = 0; kt < K; kt += KT) {
        const int buf = (kt / KT) & 1;
        // Drain our own async copies, then barrier so every wave's writes to
        // ldsB[buf] are visible (and everyone is done reading ldsB[buf^1]).
        asm volatile("s_wait_asynccnt 0x0" ::: "memory");
        __syncthreads();
        if (kt + KT < K) loadBtile(kt + KT, buf ^ 1);

        const float* __restrict__ Bt = &ldsB[buf][0];
#pragma unroll 4
        for (int k0 = 0; k0 < KT; k0 += 4) {
            const int kk = k0 + khalf * 2;
            v2f a;
            a.x = Arow[kt + kk];
            a.y = Arow[kt + kk + 1];

            const float* __restrict__ B0 = Bt + kk * 64 + nIdx;
            const float* __restrict__ B1 = B0 + 64;
            v2f b0; b0.x = B0[0];  b0.y = B1[0];
            v2f b1; b1.x = B0[16]; b1.y = B1[16];
            v2f b2; b2.x = B0[32]; b2.y = B1[32];
            v2f b3; b3.x = B0[48]; b3.y = B1[48];

            acc0 = __builtin_amdgcn_wmma_f32_16x16x4_f32(false, a, false, b0, (short)0, acc0, false, false);
            acc1 = __builtin_amdgcn_wmma_f32_16x16x4_f32(false, a, false, b1, (short)0, acc1, false, false);
            acc2 = __builtin_amdgcn_wmma_f32_16x16x4_f32(false, a, false, b2, (short)0, acc2, false, false);
            acc3 = __builtin_amdgcn_wmma_f32_16x16x4_f32(false, a, false, b3, (short)0, acc3, false, false);
        }
    }

    const int  col  = colBase + nIdx;
    const float bz0 = bias[col +  0];
    const float bz1 = bias[col + 16];
    const float bz2 = bias[col + 32];
    const float bz3 = bias[col + 48];

#pragma unroll
    for (int r = 0; r < 8; ++r) {
        const int row = rowBase + r + 8 * khalf;
        if (row < M) {
            float* __restrict__ Crow = C + (long long)row * N + col;
            float v0 = acc0[r] + bz0;
            float v1 = acc1[r] + bz1;
            float v2 = acc2[r] + bz2;
            float v3 = acc3[r] + bz3;
            if (relu) {
                v0 = fmaxf(v0, 0.0f); v1 = fmaxf(v1, 0.0f);
                v2 = fmaxf(v2, 0.0f); v3 = fmaxf(v3, 0.0f);
            }
            Crow[0] = v0; Crow[16] = v1; Crow[32] = v2; Crow[48] = v3;
        }
    }
}

// ---------------------------------------------------------------- BatchNorm

__global__ void __launch_bounds__(HID)
bn_stats(const float* __restrict__ h, float* __restrict__ sum,
         float* __restrict__ sumsq, int M, int rpb) {
    const int c  = threadIdx.x;               // channel, blockDim.x == HID
    const int r0 = blockIdx.x * rpb;
    int r1 = r0 + rpb; if (r1 > M) r1 = M;
    float s = 0.0f, s2 = 0.0f;
    for (int r = r0; r < r1; ++r) {
        float v = h[(long long)r * HID + c];
        s  += v;
        s2 += v * v;
    }
    atomAddF(&sum[c],   s);
    atomAddF(&sumsq[c], s2);
}

__global__ void __launch_bounds__(HID)
bn_finalize(const float* __restrict__ sum, const float* __restrict__ sumsq,
            const float* __restrict__ gamma, const float* __restrict__ beta,
            float* __restrict__ scale, float* __restrict__ shift, int M) {
    const int c = threadIdx.x;
    const float invM   = 1.0f / (float)M;
    const float mean   = sum[c] * invM;
    const float var    = sumsq[c] * invM - mean * mean;
    const float invstd = rsqrtf(var + BN_EPS);
    const float sc     = gamma[c] * invstd;
    scale[c] = sc;
    shift[c] = beta[c] - mean * sc;
}

__global__ void bn_apply_relu(float* __restrict__ h, const float* __restrict__ scale,
                              const float* __restrict__ shift, long long n) {
    long long i = (long long)blockIdx.x * blockDim.x + threadIdx.x;
    if (i < n) {
        const int c = (int)(i & (HID - 1));   // HID == 256, power of two
        h[i] = fmaxf(h[i] * scale[c] + shift[c], 0.0f);
    }
}

// ------------------------------------------------------------------ pooling
// batch[] is sorted -> accumulate runs locally, one atomic per run/channel.

__global__ void __launch_bounds__(HID)
pool_accum(const float* __restrict__ h, const long long* __restrict__ batch,
           float* __restrict__ psum, int M, int rpb) {
    const int c  = threadIdx.x;
    const int r0 = blockIdx.x * rpb;
    int r1 = r0 + rpb; if (r1 > M) r1 = M;
    long long cur = -1;
    float acc = 0.0f;
    for (int r = r0; r < r1; ++r) {
        long long g = batch[r];
        if (g != cur) {
            if (cur >= 0) atomAddF(&psum[cur * HID + c], acc);
            cur = g;
            acc = 0.0f;
        }
        acc += h[(long long)r * HID + c];
    }
    if (cur >= 0) atomAddF(&psum[cur * HID + c], acc);
}

__global__ void pool_count(const long long* __restrict__ batch,
                           float* __restrict__ cnt, int M) {
    int i = blockIdx.x * blockDim.x + threadIdx.x;
    if (i < M) atomAddF(&cnt[batch[i]], 1.0f);
}

__global__ void pool_finalize(const float* __restrict__ psum,
                              const float* __restrict__ cnt,
                              float* __restrict__ pooled, int n) {
    int i = blockIdx.x * blockDim.x + threadIdx.x;
    if (i < n) pooled[i] = psum[i] / fmaxf(cnt[i / HID], 1.0f);
}

// -------------------------------------------------------------------- launch

static inline unsigned cdiv(long long a, long long b) { return (unsigned)((a + b - 1) / b); }

extern "C" void kernel_launch(void* const* d_in, const int* in_sizes, int n_in,
                              void* d_out, int out_size, void* d_ws, size_t ws_size,
                              hipStream_t stream) {
    const float*     x     = (const float*)d_in[0];
    const long long* ei    = (const long long*)d_in[1];
    const long long* batch = (const long long*)d_in[2];
    const float *W1a = (const float*)d_in[3],  *b1a = (const float*)d_in[4];
    const float *W2a = (const float*)d_in[5],  *b2a = (const float*)d_in[6];
    const float *ga  = (const float*)d_in[7],  *ba  = (const float*)d_in[8];
    const float *W1b = (const float*)d_in[9],  *b1b = (const float*)d_in[10];
    const float *W2b = (const float*)d_in[11], *b2b = (const float*)d_in[12];
    const float *gb  = (const float*)d_in[13], *bb  = (const float*)d_in[14];
    const float *Wfc = (const float*)d_in[15], *bfc = (const float*)d_in[16];
    float* out = (float*)d_out;

    const int       M = in_sizes[0] / IN_C;     // 50000 nodes
    const long long E = in_sizes[1] / 2;        // 800000 edges
    const long long* esrc = ei;
    const long long* edst = ei + E;

    // -------- workspace carve-out (floats) --------
    float* buf0   = (float*)d_ws;                        // [M, HID] aggregate
    float* buf1   = buf0 + (long long)M * HID;           // [M, HID] mlp hidden
    float* buf2   = buf1 + (long long)M * HID;           // [M, HID] layer out
    float* sum    = buf2 + (long long)M * HID;           // [HID]
    float* sumsq  = sum   + HID;                         // [HID]
    float* scale  = sumsq + HID;                         // [HID]
    float* shift  = scale + HID;                         // [HID]
    float* psum   = shift + HID;                         // [NGRAPH, HID]
    float* pcnt   = psum  + (long long)NGRAPH * HID;     // [NGRAPH]
    float* pooled = pcnt  + NGRAPH;                      // [NGRAPH, HID]

    const int T = 256;
    const int RPB = 128;                                  // rows per stats block

    // zero the stats + pooling region once per call
    {
        long long nz = 4LL * HID + (long long)NGRAPH * HID + NGRAPH
                     + (long long)NGRAPH * HID;
        zero_f32<<<cdiv(nz, T), T, 0, stream>>>(sum, nz);
    }

    // =================== GIN layer A (IN_C -> HID) ===================
    copy_f32v4<<<cdiv((long long)M * IN_C / 4, T), T, 0, stream>>>(buf0, x, (long long)M * IN_C / 4);
    scatter_edges<<<cdiv(E * (IN_C / 4), T), T, 0, stream>>>(buf0, x, esrc, edst, E, IN_C);

    {
        dim3 g1(cdiv(M, 128), HID / 64);
        wmma_gemm_f32<<<g1, T, 0, stream>>>(buf0, W1a, b1a, buf1, M, IN_C, HID, 1);
        wmma_gemm_f32<<<g1, T, 0, stream>>>(buf1, W2a, b2a, buf2, M, HID,  HID, 0);
    }
    bn_stats<<<cdiv(M, RPB), HID, 0, stream>>>(buf2, sum, sumsq, M, RPB);
    bn_finalize<<<1, HID, 0, stream>>>(sum, sumsq, ga, ba, scale, shift, M);
    bn_apply_relu<<<cdiv((long long)M * HID, T), T, 0, stream>>>(buf2, scale, shift, (long long)M * HID);

    // =================== GIN layer B (HID -> HID) ====================
    zero_f32<<<cdiv(2LL * HID, T), T, 0, stream>>>(sum, 2LL * HID);
    copy_f32v4<<<cdiv((long long)M * HID / 4, T), T, 0, stream>>>(buf0, buf2, (long long)M * HID / 4);
    scatter_edges<<<cdiv(E * (HID / 4), T), T, 0, stream>>>(buf0, buf2, esrc, edst, E, HID);

    {
        dim3 g1(cdiv(M, 128), HID / 64);
        wmma_gemm_f32<<<g1, T, 0, stream>>>(buf0, W1b, b1b, buf1, M, HID, HID, 1);
        wmma_gemm_f32<<<g1, T, 0, stream>>>(buf1, W2b, b2b, buf2, M, HID, HID, 0);
    }
    bn_stats<<<cdiv(M, RPB), HID, 0, stream>>>(buf2, sum, sumsq, M, RPB);
    bn_finalize<<<1, HID, 0, stream>>>(sum, sumsq, gb, bb, scale, shift, M);
    bn_apply_relu<<<cdiv((long long)M * HID, T), T, 0, stream>>>(buf2, scale, shift, (long long)M * HID);

    // =================== global mean pool + FC =======================
    pool_count<<<cdiv(M, T), T, 0, stream>>>(batch, pcnt, M);
    pool_accum<<<cdiv(M, 256), HID, 0, stream>>>(buf2, batch, psum, M, 256);
    pool_finalize<<<cdiv(NGRAPH * HID, T), T, 0, stream>>>(psum, pcnt, pooled, NGRAPH * HID);

    {
        dim3 gf(cdiv(NGRAPH, 128), OUT_C / 64);
        wmma_gemm_f32<<<gf, T, 0, stream>>>(pooled, Wfc, bfc, out, NGRAPH, HID, OUT_C, 0);
    }
}